// EncoderModel_18674517803455
// MI455X (gfx1250) — compile-verified
//
#include <hip/hip_runtime.h>
#include <hip/hip_bf16.h>

// ---------------------------------------------------------------------------
// Sparse 3D conv encoder for MI455X (gfx1250, wave32).
// Hash-table voxel merge (L2-resident), WMMA f32 16x16x4 submanifold conv,
// fused BN+ReLU+downsample-conv, stride-2 pooling via hashed re-insert,
// final dense scatter.
// ---------------------------------------------------------------------------

typedef __attribute__((ext_vector_type(2))) float v2f;
typedef __attribute__((ext_vector_type(8))) float v8f;

#define ENC_B    32
#define ENC_C    4
#define ENC_N    1048576          // 32 * 32768 points
#define ENC_CAP  (1u << 21)       // hash capacity (2x max unique)
#define ENC_EMPTY 0xFFFFFFFFFFFFFFFFull
#define ENC_EPS  1e-4f

// ---------------------------------------------------------------------------
// hash-table helpers (open addressing, linear probe; CAP is power of two)
// ---------------------------------------------------------------------------
__device__ __forceinline__ unsigned enc_h0(unsigned long long key) {
    return (unsigned)((key * 0x9E3779B97F4A7C15ull) >> 43) & (ENC_CAP - 1u);
}

__device__ __forceinline__ int enc_insert(unsigned long long* keys,
                                          unsigned long long key,
                                          int* list, int* counter) {
    unsigned h = enc_h0(key);
    for (;;) {
        unsigned long long prev = atomicCAS(&keys[h], ENC_EMPTY, key);
        if (prev == ENC_EMPTY) {           // claimed a fresh slot
            int idx = atomicAdd(counter, 1);
            list[idx] = (int)h;
            return (int)h;
        }
        if (prev == key) return (int)h;    // already present
        h = (h + 1u) & (ENC_CAP - 1u);
    }
}

__device__ __forceinline__ int enc_lookup(const unsigned long long* keys,
                                          unsigned long long key) {
    unsigned h = enc_h0(key);
    for (;;) {
        unsigned long long k = keys[h];
        if (k == key) return (int)h;
        if (k == ENC_EMPTY) return -1;
        h = (h + 1u) & (ENC_CAP - 1u);
    }
}

// ---------------------------------------------------------------------------
// table init: keys <- EMPTY, feats <- 0, counter <- 0
// ---------------------------------------------------------------------------
__global__ void enc_init_table(unsigned long long* __restrict__ keys,
                               float* __restrict__ feat,
                               int* __restrict__ counter) {
    unsigned i = blockIdx.x * blockDim.x + threadIdx.x;
    if (i < ENC_CAP) {
        keys[i] = ENC_EMPTY;
        feat[i * 4 + 0] = 0.f; feat[i * 4 + 1] = 0.f;
        feat[i * 4 + 2] = 0.f; feat[i * 4 + 3] = 0.f;
        if (i == 0) *counter = 0;
    }
}

__global__ void enc_zero_stats(float* __restrict__ stats) {
    if (threadIdx.x < 8) stats[threadIdx.x] = 0.f;
}

__global__ void enc_zero(float* __restrict__ p, int n) {
    int i = blockIdx.x * blockDim.x + threadIdx.x;
    if (i < n) p[i] = 0.f;
}

// ---------------------------------------------------------------------------
// level-0 insert + duplicate merge (segment_sum equivalent, order-free)
// key = ((b*1024 + x)*1024 + y)*1024 + z  -> pure bitfield, decodable
// ---------------------------------------------------------------------------
__global__ void enc_insert_points(const float* __restrict__ pc,
                                  unsigned long long* __restrict__ keys,
                                  float* __restrict__ feat,
                                  int* __restrict__ list,
                                  int* __restrict__ counter) {
    int i = blockIdx.x * blockDim.x + threadIdx.x;
    if (i >= ENC_N) return;
    const float* p = pc + (size_t)i * 5;
    unsigned long long x = (unsigned long long)(long long)p[0];
    unsigned long long y = (unsigned long long)(long long)p[1];
    unsigned long long z = (unsigned long long)(long long)p[2];
    unsigned long long b = (unsigned long long)(long long)p[3];
    float f = p[4];
    unsigned long long key = (((((b << 10) | x) << 10) | y) << 10) | z;
    int h = enc_insert(keys, key, list, counter);
    atomicAdd(&feat[h * 4 + 0], f);
}

// ---------------------------------------------------------------------------
// submanifold 3x3x3 conv, 1 -> 4 channels, via V_WMMA_F32_16X16X4_F32.
// One wave handles a tile of 16 active voxels:
//   gather 27 neighbor scalars/voxel into LDS (point-major, K padded to 32),
//   then 7 WMMA k-steps accumulate G(16x28) x W(28x16-padded).
// A layout: lane=row M; VGPR0/1 hold K{0,1} (lanes 0-15) / K{2,3} (16-31)
//   -> per step one aligned ds_load_b64 per lane.
// B fragments are tile-invariant: preloaded once per wave, branchlessly
//   (clamped index + select, no exec-mask divergence in the MMA chain).
// C/D: VGPR v holds M=v (lanes 0-15) / M=v+8 (lanes 16-31).
// ---------------------------------------------------------------------------
__global__ void __launch_bounds__(256)
enc_subm_conv(const unsigned long long* __restrict__ keys,
              const float* __restrict__ feat,          // channel 0 = merged scalar
              const int* __restrict__ list,
              const int* __restrict__ counter,
              const float* __restrict__ Wsub,          // (27,1,4)
              float* __restrict__ outFeat) {           // (CAP,4)
    __shared__ float gsh[8][16 * 32];                  // 8 waves, g[row][K], K pad 32
    const int lane = threadIdx.x & 31;
    const int wave = threadIdx.x >> 5;
    const int tile = blockIdx.x * (blockDim.x >> 5) + wave;
    float* g = gsh[wave];

    const int row  = lane & 15;
    const int kofs = (lane < 16) ? 0 : 2;

    // ---- preload tile-invariant B fragments (7 x v2f), branch-free --------
    v2f bfrag[7];
#pragma unroll
    for (int kk = 0; kk < 7; ++kk) {
        int kb  = kk * 4 + kofs;
        int kc0 = (kb     < 27) ? kb     : 26;
        int kc1 = (kb + 1 < 27) ? kb + 1 : 26;
        float w0 = Wsub[kc0 * ENC_C + (row & 3)];      // unconditional load
        float w1 = Wsub[kc1 * ENC_C + (row & 3)];
        bool okc = (row < ENC_C);
        bfrag[kk].x = (okc && kb     < 27) ? w0 : 0.f; // v_cndmask, no branch
        bfrag[kk].y = (okc && kb + 1 < 27) ? w1 : 0.f;
    }

    const int cnt = *counter;
    const int tileBase = tile * 16;
    const bool active = tileBase < cnt;                // wave-uniform

    if (active) __builtin_prefetch(&list[tileBase], 0, 0);

    // ---- gather phase: 16 rows x 32 K-slots = 512 = 16*32 LDS entries -----
    for (int j = 0; j < 16; ++j) {
        int idx = j * 32 + lane;
        int m = idx >> 5, k = idx & 31;
        float val = 0.f;
        if (active && k < 27 && (tileBase + m) < cnt) {
            int slot = list[tileBase + m];
            unsigned long long key = keys[slot];
            int z = (int)(key & 1023u);
            int y = (int)((key >> 10) & 1023u);
            int x = (int)((key >> 20) & 1023u);
            unsigned long long b = key >> 30;
            int dx = k / 9 - 1, dy = (k / 3) % 3 - 1, dz = k % 3 - 1;
            int nx = x + dx, ny = y + dy, nz = z + dz;
            if ((unsigned)nx < 1024u && (unsigned)ny < 1024u && (unsigned)nz < 1024u) {
                unsigned long long qk =
                    (((((b << 10) | (unsigned)nx) << 10) | (unsigned)ny) << 10) | (unsigned)nz;
                int h = enc_lookup(keys, qk);
                if (h >= 0) val = feat[h * 4 + 0];
            }
        }
        g[m * 32 + k] = val;                           // k in [27,32) stays 0
    }
    __syncthreads();

    if (active) {                                      // EXEC all-ones for WMMA
        // load all A fragments first: 7 aligned ds_load_b64, single wait
        v2f afrag[7];
#pragma unroll
        for (int kk = 0; kk < 7; ++kk)
            afrag[kk] = *(const v2f*)&g[row * 32 + kk * 4 + kofs];

        v8f c = {};
#pragma unroll
        for (int kk = 0; kk < 7; ++kk)
            c = __builtin_amdgcn_wmma_f32_16x16x4_f32(
                    false, afrag[kk], false, bfrag[kk], (short)0, c, false, false);

        // write back: column (N) = out channel; rows M = tile points
        if (row < ENC_C) {
            int mbase = (lane >= 16) ? 8 : 0;
#pragma unroll
            for (int v = 0; v < 8; ++v) {
                int m = mbase + v;
                if (tileBase + m < cnt) {
                    int slot = list[tileBase + m];
                    outFeat[slot * 4 + row] = c[v];
                }
            }
        }
    }
}

// ---------------------------------------------------------------------------
// per-channel sum / sum-of-squares over active voxels (for BN)
// ---------------------------------------------------------------------------
__global__ void enc_stats(const float* __restrict__ feat,
                          const int* __restrict__ list,
                          const int* __restrict__ counter,
                          float* __restrict__ stats) {   // [0..3]=sum, [4..7]=sumsq
    __shared__ float ssum[4], ssq[4];
    if (threadIdx.x < 4) { ssum[threadIdx.x] = 0.f; ssq[threadIdx.x] = 0.f; }
    __syncthreads();
    int cnt = *counter;
    int i = blockIdx.x * blockDim.x + threadIdx.x;
    if (i < cnt) {
        int slot = list[i];
#pragma unroll
        for (int ch = 0; ch < 4; ++ch) {
            float f = feat[slot * 4 + ch];
            atomicAdd(&ssum[ch], f);
            atomicAdd(&ssq[ch], f * f);
        }
    }
    __syncthreads();
    if (threadIdx.x < 4) {
        atomicAdd(&stats[threadIdx.x], ssum[threadIdx.x]);
        atomicAdd(&stats[4 + threadIdx.x], ssq[threadIdx.x]);
    }
}

// ---------------------------------------------------------------------------
// downsample pass. mode 0: BN+ReLU -> W_down[koff] 4x4 matvec -> insert@S/2
//                  mode 1: feat/8                           -> insert@S/2
// ---------------------------------------------------------------------------
__global__ void enc_downsample(const unsigned long long* __restrict__ keysSrc,
                               const float* __restrict__ featSrc,
                               const int* __restrict__ listSrc,
                               const int* __restrict__ cntSrc,
                               unsigned long long* __restrict__ keysDst,
                               float* __restrict__ featDst,
                               int* __restrict__ listDst,
                               int* __restrict__ cntDst,
                               int srcSb, int mode,
                               const float* __restrict__ stats,
                               const float* __restrict__ gamma,
                               const float* __restrict__ beta,
                               const float* __restrict__ Wd) {  // (8,4,4)
    int cnt = *cntSrc;
    int i = blockIdx.x * blockDim.x + threadIdx.x;
    if (i >= cnt) return;

    int slot = listSrc[i];
    unsigned long long key = keysSrc[slot];
    unsigned mask = (1u << srcSb) - 1u;
    int z = (int)(key & mask);
    int y = (int)((key >> srcSb) & mask);
    int x = (int)((key >> (2 * srcSb)) & mask);
    unsigned long long b = key >> (3 * srcSb);

    float f[4], o[4];
#pragma unroll
    for (int ch = 0; ch < 4; ++ch) f[ch] = featSrc[slot * 4 + ch];

    if (mode == 0) {
        float cntf = (cnt > 0) ? (float)cnt : 1.f;
        float yv[4];
#pragma unroll
        for (int ch = 0; ch < 4; ++ch) {
            float mean = stats[ch] / cntf;
            float var  = stats[4 + ch] / cntf - mean * mean;
            float nrm  = (f[ch] - mean) * rsqrtf(var + ENC_EPS);
            yv[ch] = fmaxf(nrm * gamma[ch] + beta[ch], 0.f);
        }
        int koff = (x & 1) * 4 + (y & 1) * 2 + (z & 1);
        const float* W = Wd + koff * 16;
#pragma unroll
        for (int oc = 0; oc < 4; ++oc)
            o[oc] = yv[0] * W[oc] + yv[1] * W[4 + oc]
                  + yv[2] * W[8 + oc] + yv[3] * W[12 + oc];
    } else {
#pragma unroll
        for (int ch = 0; ch < 4; ++ch) o[ch] = f[ch] * 0.125f;
    }

    int nSb = srcSb - 1;
    unsigned long long nkey =
        (((((b << nSb) | (unsigned)(x >> 1)) << nSb) | (unsigned)(y >> 1)) << nSb)
        | (unsigned)(z >> 1);
    int h = enc_insert(keysDst, nkey, listDst, cntDst);
#pragma unroll
    for (int ch = 0; ch < 4; ++ch) atomicAdd(&featDst[h * 4 + ch], o[ch]);
}

// ---------------------------------------------------------------------------
// scatter final S=16 level into dense (B, C, 16, 16, 16)
// ---------------------------------------------------------------------------
__global__ void enc_scatter(const unsigned long long* __restrict__ keys,
                            const float* __restrict__ feat,
                            const int* __restrict__ list,
                            const int* __restrict__ counter,
                            float* __restrict__ out) {
    int cnt = *counter;
    int i = blockIdx.x * blockDim.x + threadIdx.x;
    if (i >= cnt) return;
    int slot = list[i];
    unsigned long long key = keys[slot];
    int z = (int)(key & 15u);
    int y = (int)((key >> 4) & 15u);
    int x = (int)((key >> 8) & 15u);
    int b = (int)(key >> 12);
    if (b >= ENC_B) b = ENC_B - 1;
#pragma unroll
    for (int c = 0; c < 4; ++c) {
        size_t idx = ((((size_t)b * ENC_C + c) * 16 + x) * 16 + y) * 16 + z;
        out[idx] = feat[slot * 4 + c];   // keys unique -> plain store into zeroed grid
    }
}

// ---------------------------------------------------------------------------
// launcher
// ---------------------------------------------------------------------------
extern "C" void kernel_launch(void* const* d_in, const int* in_sizes, int n_in,
                              void* d_out, int out_size, void* d_ws, size_t ws_size,
                              hipStream_t stream) {
    const float* pc    = (const float*)d_in[0];   // (N,5)
    const float* Wsub  = (const float*)d_in[1];   // (27,1,4)
    const float* Wdown = (const float*)d_in[2];   // (3,8,4,4)
    const float* gamma = (const float*)d_in[3];   // (3,4)
    const float* beta  = (const float*)d_in[4];   // (3,4)
    float* out = (float*)d_out;

    const size_t CAP = ENC_CAP;
    char* ws = (char*)d_ws;
    const size_t need = 72 * CAP + 256;
    if (ws_size < need) return;

    unsigned long long* keysA = (unsigned long long*)(ws + 0 * CAP);
    unsigned long long* keysB = (unsigned long long*)(ws + 8 * CAP);
    float* featA = (float*)(ws + 16 * CAP);
    float* featB = (float*)(ws + 32 * CAP);
    float* featC = (float*)(ws + 48 * CAP);      // submanifold-conv output
    int*   listA = (int*)(ws + 64 * CAP);
    int*   listB = (int*)(ws + 68 * CAP);
    int*   cntA  = (int*)(ws + 72 * CAP);
    int*   cntB  = cntA + 1;
    float* stats = (float*)(ws + 72 * CAP + 16);

    const int TB = 256;
    const int gCap  = (int)(CAP / TB);           // table init
    const int gPts  = ENC_N / TB;                // point/list passes (upper bound)
    const int gConv = (int)(CAP / 16 / 8);       // 16-pt tiles, 8 waves/block

    // level 0: hash-merge points
    enc_init_table<<<gCap, TB, 0, stream>>>(keysA, featA, cntA);
    enc_insert_points<<<gPts, TB, 0, stream>>>(pc, keysA, featA, listA, cntA);

    // submanifold conv (WMMA) -> featC
    enc_subm_conv<<<gConv, TB, 0, stream>>>(keysA, featA, listA, cntA, Wsub, featC);

    const float* curFeat = featC;
    int srcSb = 10;                              // S = 1024
    for (int it = 0; it < 3; ++it) {
        enc_zero_stats<<<1, 8, 0, stream>>>(stats);
        enc_stats<<<gPts, TB, 0, stream>>>(curFeat, listA, cntA, stats);

        enc_init_table<<<gCap, TB, 0, stream>>>(keysB, featB, cntB);
        enc_downsample<<<gPts, TB, 0, stream>>>(keysA, curFeat, listA, cntA,
                                                keysB, featB, listB, cntB,
                                                srcSb, 0, stats,
                                                gamma + it * 4, beta + it * 4,
                                                Wdown + it * 128);
        srcSb -= 1;

        enc_init_table<<<gCap, TB, 0, stream>>>(keysA, featA, cntA);
        enc_downsample<<<gPts, TB, 0, stream>>>(keysB, featB, listB, cntB,
                                                keysA, featA, listA, cntA,
                                                srcSb, 1, nullptr, nullptr,
                                                nullptr, nullptr);
        srcSb -= 1;
        curFeat = featA;
    }

    // dense scatter (S = 16)
    enc_zero<<<(out_size + TB - 1) / TB, TB, 0, stream>>>(out, out_size);
    enc_scatter<<<gPts, TB, 0, stream>>>(keysA, featA, listA, cntA, out);
}